// DRMM_56238301773937
// MI455X (gfx1250) — compile-verified
//
#include <hip/hip_runtime.h>
#include <hip/hip_bf16.h>
#include <math.h>

// ---------------------------------------------------------------------------
// DRMM on gfx1250 (MI455X).
//   Kernel 1: zero workspace histogram.
//   Kernel 2: cosine-sim via V_WMMA_F32_16X16X4_F32 + histogram (LDS -> global atomics).
//   Kernel 3: log1p + tanh-MLP + softmax gate + weighted sum  (1 wave per batch).
// ---------------------------------------------------------------------------

typedef __attribute__((ext_vector_type(2))) float v2f;
typedef __attribute__((ext_vector_type(8))) float v8f;

#define VV     50000
#define EE     300
#define BB     32
#define QQ     16
#define DD     2048
#define NB     30
#define EPSF   1e-8f

__global__ __launch_bounds__(256) void drmm_zero_hist(unsigned int* __restrict__ h, int n) {
    int i = blockIdx.x * blockDim.x + threadIdx.x;
    if (i < n) h[i] = 0u;
}

// grid = B * (D/128) blocks, 256 threads (8 waves). Each wave: one 16x16 sim tile.
__global__ __launch_bounds__(256)
void drmm_sim_hist(const int* __restrict__ query,
                   const int* __restrict__ document,
                   const float* __restrict__ emb,
                   unsigned int* __restrict__ hist)
{
    __shared__ float        sQ[QQ * EE];       // 16 x 300 query embeddings (stride 300: bank-conflict-free)
    __shared__ float        sQn[QQ];           // query L2 norms
    __shared__ unsigned int sHist[QQ * NB];    // per-block partial histogram

    const int b       = blockIdx.x >> 4;       // 16 blocks per batch
    const int tileBlk = blockIdx.x & 15;       // 128 docs per block
    const int tid     = threadIdx.x;

    // Stage the 16 query-embedding rows into LDS (row gather, contiguous within row).
    for (int idx = tid; idx < QQ * EE; idx += 256) {
        int m = idx / EE;
        int k = idx - m * EE;
        long long row = (long long)query[b * QQ + m] * EE;
        sQ[idx] = emb[row + k];
    }
    for (int i = tid; i < QQ * NB; i += 256) sHist[i] = 0u;
    __syncthreads();

    if (tid < QQ) {
        float ss = 0.f;
        for (int k = 0; k < EE; ++k) { float v = sQ[tid * EE + k]; ss = fmaf(v, v, ss); }
        sQn[tid] = sqrtf(ss);
    }
    __syncthreads();

    const int wave = tid >> 5;
    const int lane = tid & 31;
    const int n    = lane & 15;                // doc column within tile == A row M for this lane
    const int hi   = lane >> 4;                // half-wave selector (K sub-pair / C row offset)
    const int d0   = tileBlk * 128 + wave * 16;

    const long long docRow = (long long)document[b * DD + d0 + n] * EE;

    // Operand base pointers. Per ISA 16x4 f32 A layout:
    //   lane L: M = L%16, VGPR0 = K(kb + 2*(L/16)), VGPR1 = K(+1)   -> one v2f load.
    const float* __restrict__ aBase = &sQ[n * EE + 2 * hi];
    const float* __restrict__ bBase = &emb[docRow + 2 * hi];

    v8f  c   = {0.f, 0.f, 0.f, 0.f, 0.f, 0.f, 0.f, 0.f};
    float ssq = 0.f;                           // half of doc n's sum-of-squares lives in this lane

    #pragma unroll 5
    for (int kb = 0; kb < EE; kb += 4) {       // 75 iterations, K = 300
        v2f a  = *(const v2f*)(aBase + kb);    // LDS,    8B aligned
        v2f bv = *(const v2f*)(bBase + kb);    // global, 8B aligned (L2-resident table)
        ssq = fmaf(bv.x, bv.x, ssq);
        ssq = fmaf(bv.y, bv.y, ssq);
        // D = A(16x4 f32) * B(4x16 f32) + C(16x16 f32)
        c = __builtin_amdgcn_wmma_f32_16x16x4_f32(false, a, false, bv,
                                                  (short)0, c, false, false);
    }

    // Doc norm: lanes n and n^16 hold complementary K subsets.
    float dn = sqrtf(ssq + __shfl_xor(ssq, 16, 32));

    // C layout: VGPR i of lane L -> (M = i + 8*(L/16), N = L%16).
    #pragma unroll
    for (int i = 0; i < 8; ++i) {
        int   m     = i + 8 * hi;
        float denom = fmaxf(sQn[m] * dn, EPSF);
        float sim   = c[i] / denom;
        int   bin   = (int)floorf((sim + 1.0f) * (NB * 0.5f));
        bin = bin < 0 ? 0 : (bin > NB - 1 ? NB - 1 : bin);
        atomicAdd(&sHist[m * NB + bin], 1u);
    }
    __syncthreads();

    for (int i = tid; i < QQ * NB; i += 256) {
        unsigned int v = sHist[i];
        if (v) atomicAdd(&hist[b * QQ * NB + i], v);
    }
}

// One wave per batch: gate dot, MLP, softmax, weighted sum.
__global__ __launch_bounds__(32)
void drmm_mlp(const int* __restrict__ query,
              const int* __restrict__ qlen,
              const float* __restrict__ emb,
              const unsigned int* __restrict__ hist,
              const float* __restrict__ w1, const float* __restrict__ b1,
              const float* __restrict__ w2, const float* __restrict__ b2,
              const float* __restrict__ w3, const float* __restrict__ b3,
              const float* __restrict__ gw, const float* __restrict__ gb,
              float* __restrict__ out)
{
    const int b    = blockIdx.x;
    const int lane = threadIdx.x;

    float s  = -INFINITY;   // gate logit
    float z3 = 0.f;

    if (lane < QQ) {
        const int q = lane;
        long long row = (long long)query[b * QQ + q] * EE;
        float acc = gb[0];
        for (int k = 0; k < EE; ++k) acc = fmaf(emb[row + k], gw[k], acc);
        s = acc;

        const bool active = q < qlen[b];
        float h[NB];
        #pragma unroll
        for (int i = 0; i < NB; ++i)
            h[i] = active ? log1pf((float)hist[(b * QQ + q) * NB + i]) : 0.f;

        float z2 = b2[0];
        #pragma unroll
        for (int j = 0; j < 5; ++j) {
            float z1 = b1[j];
            #pragma unroll
            for (int i = 0; i < NB; ++i) z1 = fmaf(h[i], w1[i * 5 + j], z1);
            z2 = fmaf(tanhf(z1), w2[j], z2);
        }
        z3 = tanhf(fmaf(tanhf(z2), w3[0], b3[0]));
    }

    // softmax over lanes 0..15 (others contribute exp(-inf)=0)
    float mx = s;
    for (int off = 16; off >= 1; off >>= 1) mx = fmaxf(mx, __shfl_xor(mx, off, 32));
    float e = (lane < QQ) ? expf(s - mx) : 0.f;
    float denom = e;
    float num   = e * z3;
    for (int off = 16; off >= 1; off >>= 1) {
        denom += __shfl_xor(denom, off, 32);
        num   += __shfl_xor(num,   off, 32);
    }
    if (lane == 0) out[b] = num / denom;
}

extern "C" void kernel_launch(void* const* d_in, const int* in_sizes, int n_in,
                              void* d_out, int out_size, void* d_ws, size_t ws_size,
                              hipStream_t stream) {
    (void)in_sizes; (void)n_in; (void)out_size; (void)ws_size;

    const int*   query    = (const int*)  d_in[0];   // [B,Q]
    const int*   qlen     = (const int*)  d_in[1];   // [B]
    const int*   document = (const int*)  d_in[2];   // [B,D]
    const float* emb      = (const float*)d_in[3];   // [V,E]
    const float* w1       = (const float*)d_in[4];   // [30,5]
    const float* b1       = (const float*)d_in[5];   // [5]
    const float* w2       = (const float*)d_in[6];   // [5,1]
    const float* b2       = (const float*)d_in[7];   // [1]
    const float* w3       = (const float*)d_in[8];   // [1,1]
    const float* b3       = (const float*)d_in[9];   // [1]
    const float* gw       = (const float*)d_in[10];  // [E,1]
    const float* gb       = (const float*)d_in[11];  // [1]

    unsigned int* hist = (unsigned int*)d_ws;        // [B,Q,NB] = 15360 uints
    const int nh = BB * QQ * NB;

    drmm_zero_hist<<<(nh + 255) / 256, 256, 0, stream>>>(hist, nh);
    drmm_sim_hist<<<BB * (DD / 128), 256, 0, stream>>>(query, document, emb, hist);
    drmm_mlp<<<BB, 32, 0, stream>>>(query, qlen, emb, hist,
                                    w1, b1, w2, b2, w3, b3, gw, gb,
                                    (float*)d_out);
}